// VisionAttention_24936580120641
// MI455X (gfx1250) — compile-verified
//
#include <hip/hip_runtime.h>
#include <hip/hip_bf16.h>

typedef __attribute__((ext_vector_type(2))) float v2f;
typedef __attribute__((ext_vector_type(8))) float v8f;
typedef __attribute__((address_space(3))) float lds_float;

#define DIM 1280
#define QKV_DIM 3840
#define NUM_VISUAL 16
#define MAX_PATCH 1024
#define NUM_HEADS 16
#define HEAD_DIM 80
#define ROWS (NUM_VISUAL * MAX_PATCH)   // 16384

#define BM 64
#define BN 64
#define BK 32
#define CHUNK_FLOATS (BM * BK)          // 2048 floats per A (and per B) chunk

// ---------------------------------------------------------------------------
// LDS-tiled GEMM: C[M x N] = A[M x K] * W^T (W is [N x K] row-major) + bias[N]
// Block: 256 threads (8 waves). Tile BM=64 x BN=64, K stepped by BK=32.
// A/B chunks staged into LDS with GLOBAL_LOAD_ASYNC_TO_LDS_B128 (ASYNCcnt),
// double-buffered. LDS dest passed as addrspace(3) pointer so the shared
// buffer escapes into the asm and the ds_loads are not folded to undef.
// ---------------------------------------------------------------------------
__global__ void gemm_wmma_lds(const float* __restrict__ A,
                              const float* __restrict__ W,
                              const float* __restrict__ bias,
                              float* __restrict__ C,
                              int M, int N, int K) {
    // [buf 0|1] x [ A: 2048 floats | B: 2048 floats ]  = 32 KB
    __shared__ float smem[2 * 2 * CHUNK_FLOATS];
    lds_float* sb = (lds_float*)smem;

    const int tid  = threadIdx.x;
    const int lane = tid & 31;
    const int wave = tid >> 5;
    const int l16  = lane & 15;
    const int half = lane >> 4;

    const int blockM = blockIdx.y * BM;
    const int blockN = blockIdx.x * BN;

    const int mi  = wave & 3;          // M-tile 0..3
    const int njb = (wave >> 2) * 2;   // N-tile base: 0 or 2

    // Issue one BK-chunk of A and W into LDS buffer `buf` (4 async per wave).
    // Flattened as 512 16-byte units per matrix: e -> row=e>>3, colgrp=e&7.
    auto issue_chunk = [&](int kbase, int buf) {
#pragma unroll
        for (int j = 0; j < 2; ++j) {
            const int e   = j * 256 + tid;
            const int row = e >> 3;
            const int cg  = e & 7;
            const float* g = A + (size_t)(blockM + row) * K + kbase + cg * 4;
            lds_float* dst = sb + buf * 2 * CHUNK_FLOATS + row * BK + cg * 4;
            asm volatile("global_load_async_to_lds_b128 %0, %1, off"
                         :: "v"(dst), "v"(g) : "memory");
        }
#pragma unroll
        for (int j = 0; j < 2; ++j) {
            const int e   = j * 256 + tid;
            const int row = e >> 3;
            const int cg  = e & 7;
            const float* g = W + (size_t)(blockN + row) * K + kbase + cg * 4;
            lds_float* dst = sb + buf * 2 * CHUNK_FLOATS + CHUNK_FLOATS +
                             row * BK + cg * 4;
            asm volatile("global_load_async_to_lds_b128 %0, %1, off"
                         :: "v"(dst), "v"(g) : "memory");
        }
    };

    v8f acc0 = {}, acc1 = {};
    const int nchunks = K / BK;

    issue_chunk(0, 0);
    for (int c = 0; c < nchunks; ++c) {
        const int buf = c & 1;
        if (c + 1 < nchunks) {
            issue_chunk((c + 1) * BK, buf ^ 1);
            // <=4 outstanding => chunk c (issued earlier, in-order) complete.
            asm volatile("s_wait_asynccnt 0x4" ::: "memory");
        } else {
            asm volatile("s_wait_asynccnt 0x0" ::: "memory");
        }
        __syncthreads();   // chunk c visible to all waves

        const float* sA = smem + buf * 2 * CHUNK_FLOATS;
        const float* sB = sA + CHUNK_FLOATS;
#pragma unroll
        for (int kk = 0; kk < BK; kk += 4) {
            v2f a  = *(const v2f*)(sA + (mi * 16 + l16) * BK + kk + 2 * half);
            v2f b0 = *(const v2f*)(sB + (njb * 16 + l16) * BK + kk + 2 * half);
            v2f b1 = *(const v2f*)(sB + ((njb + 1) * 16 + l16) * BK + kk + 2 * half);
            acc0 = __builtin_amdgcn_wmma_f32_16x16x4_f32(false, a, false, b0,
                                                         (short)0, acc0, false, false);
            acc1 = __builtin_amdgcn_wmma_f32_16x16x4_f32(false, a, false, b1,
                                                         (short)0, acc1, false, false);
        }
        __syncthreads();   // reads done before next chunk overwrites this buf
    }

    const float bv0 = bias[blockN + njb * 16 + l16];
    const float bv1 = bias[blockN + (njb + 1) * 16 + l16];
#pragma unroll
    for (int i = 0; i < 8; ++i) {
        const int r = blockM + mi * 16 + i + 8 * half;
        C[(size_t)r * N + blockN + njb * 16 + l16]       = acc0[i] + bv0;
        C[(size_t)r * N + blockN + (njb + 1) * 16 + l16] = acc1[i] + bv1;
    }
}

// ---------------------------------------------------------------------------
// In-place RoPE on the Q and K slices of the qkv buffer.
// qkv layout: [16384 rows][3840]: cols [0,1280)=Q, [1280,2560)=K, [2560,3840)=V
// rope_cache: [16384][160] = cos(0..79) || sin(0..79)
// ---------------------------------------------------------------------------
__global__ void rope_kernel(float* __restrict__ qkv,
                            const float* __restrict__ rope) {
    const int np = blockIdx.x;                 // flat (n, p)
    const float* rc = rope + (size_t)np * 160;
    float* row = qkv + (size_t)np * QKV_DIM;

    for (int t = threadIdx.x; t < NUM_HEADS * (HEAD_DIM / 2); t += blockDim.x) {
        const int h = t / (HEAD_DIM / 2);
        const int d = t % (HEAD_DIM / 2);
        const float c1 = rc[d],      c2 = rc[d + 40];
        const float s1 = rc[80 + d], s2 = rc[120 + d];

        float* q = row + h * HEAD_DIM;
        float x1 = q[d], x2 = q[d + 40];
        q[d]      = x1 * c1 - x2 * s1;
        q[d + 40] = x2 * c2 + x1 * s2;

        float* k = row + DIM + h * HEAD_DIM;
        x1 = k[d]; x2 = k[d + 40];
        k[d]      = x1 * c1 - x2 * s1;
        k[d + 40] = x2 * c2 + x1 * s2;
    }
}

// ---------------------------------------------------------------------------
// Flash attention, fp32 WMMA. Grid: (qtile=16, head=16, image=16), block=128
// (4 waves x 16 query rows). Per 16-key tile: K and V tiles (16x80 each) are
// staged ONCE per block into LDS via async-to-LDS (4 waves share them), then
// S = Q K^T (20 wmma), online softmax (width-16 shuffles), P via LDS
// transpose, O += P V (20 wmma) — all B-fragments come from LDS.
// ---------------------------------------------------------------------------
__global__ void flash_attn(const float* __restrict__ qkv,
                           const int* __restrict__ seq_lens,
                           float* __restrict__ out) {
    const int n = blockIdx.z;
    const int h = blockIdx.y;
    const int lane = threadIdx.x & 31;
    const int wave = threadIdx.x >> 5;
    const int l16  = lane & 15;
    const int half = lane >> 4;
    const int qRow = blockIdx.x * 64 + wave * 16;

    const int seqLen = seq_lens[n];
    const float* base = qkv + (size_t)n * MAX_PATCH * QKV_DIM;
    const float* Qp = base + h * HEAD_DIM;                 // Q slice
    const float* Kp = base + DIM + h * HEAD_DIM;           // K slice
    const float* Vp = base + 2 * DIM + h * HEAD_DIM;       // V slice

    __shared__ float ldsK[16 * HEAD_DIM];   // 5 KB: key tile, [key][dim]
    __shared__ float ldsV[16 * HEAD_DIM];   // 5 KB: val tile, [key][dim]
    __shared__ float ldsP[4 * 256];         // 4 KB: per-wave P transpose slab
    lds_float* kb = (lds_float*)ldsK;
    lds_float* vb = (lds_float*)ldsV;
    float* myP = ldsP + wave * 256;

    // Load Q fragments (16 rows x 80 dims => 20 A-frags), pre-scaled.
    const float scale = 0.11180339887498948f;              // 1/sqrt(80)
    v2f qa[20];
    {
        const float* qptr = Qp + (size_t)(qRow + l16) * QKV_DIM;
#pragma unroll
        for (int kk = 0; kk < 20; ++kk) {
            v2f t = *(const v2f*)(qptr + kk * 4 + 2 * half);
            qa[kk] = t * scale;
        }
    }

    v8f acc[5] = {};
    float m_i[8], l_i[8];
#pragma unroll
    for (int i = 0; i < 8; ++i) { m_i[i] = -1e30f; l_i[i] = 0.0f; }

    for (int kt = 0; kt < MAX_PATCH / 16; ++kt) {
        const int keyBase = kt * 16;

        __syncthreads();   // previous iteration's LDS reads complete

        // ---- stage K and V tiles (2 x 16 x 80 f32 = 640 float4 units) ----
        // 5 async b128 per thread, uniform across all 4 waves.
#pragma unroll
        for (int j = 0; j < 5; ++j) {
            const int u   = j * 128 + threadIdx.x;      // 0..639
            const int isV = (u >= 320);
            const int uu  = isV ? u - 320 : u;
            const int row = uu / 20;                    // key within tile
            const int cg  = uu % 20;                    // float4 group in dim
            const float* g = (isV ? Vp : Kp) +
                             (size_t)(keyBase + row) * QKV_DIM + cg * 4;
            lds_float* dst = (isV ? vb : kb) + row * HEAD_DIM + cg * 4;
            asm volatile("global_load_async_to_lds_b128 %0, %1, off"
                         :: "v"(dst), "v"(g) : "memory");
        }
        asm volatile("s_wait_asynccnt 0x0" ::: "memory");
        __syncthreads();   // K/V tiles visible to all waves

        // ---- S = Q * K^T  (16q x 16k), accumulate over 80 dims ----
        v8f s = {};
        {
            const float* krow = ldsK + l16 * HEAD_DIM;  // key = l16
#pragma unroll
            for (int kk = 0; kk < 20; ++kk) {
                v2f b = *(const v2f*)(krow + kk * 4 + 2 * half);
                s = __builtin_amdgcn_wmma_f32_16x16x4_f32(false, qa[kk], false, b,
                                                          (short)0, s, false, false);
            }
        }

        // ---- mask + online softmax (key index == l16 in C layout) ----
        const bool valid = (keyBase + l16) < seqLen;
        float pv[8], alpha[8];
#pragma unroll
        for (int i = 0; i < 8; ++i) {
            float si = valid ? s[i] : -1e30f;
            float rm = si;
#pragma unroll
            for (int off = 1; off < 16; off <<= 1)
                rm = fmaxf(rm, __shfl_xor(rm, off, 16));
            const float mnew = fmaxf(m_i[i], rm);
            const float a = __expf(m_i[i] - mnew);
            const float p = __expf(si - mnew);
            float rs = p;
#pragma unroll
            for (int off = 1; off < 16; off <<= 1)
                rs += __shfl_xor(rs, off, 16);
            l_i[i] = l_i[i] * a + rs;
            m_i[i] = mnew;
            alpha[i] = a;
            pv[i] = p;
        }
#pragma unroll
        for (int t = 0; t < 5; ++t)
#pragma unroll
            for (int i = 0; i < 8; ++i) acc[t][i] *= alpha[i];

        // ---- stage P through LDS: C-layout -> row-major 16x16 ----
#pragma unroll
        for (int i = 0; i < 8; ++i)
            myP[(i + 8 * half) * 16 + l16] = pv[i];
        __syncthreads();

        // ---- O += P * V  (16q x 16k) * (16k x 80d), 5 d-tiles ----
        for (int kc = 0; kc < 16; kc += 4) {
            v2f a;
            a.x = myP[l16 * 16 + kc + 2 * half];
            a.y = myP[l16 * 16 + kc + 2 * half + 1];
            const float* vrow = ldsV + (kc + 2 * half) * HEAD_DIM;
#pragma unroll
            for (int t = 0; t < 5; ++t) {
                v2f b;
                b.x = vrow[t * 16 + l16];
                b.y = vrow[HEAD_DIM + t * 16 + l16];
                acc[t] = __builtin_amdgcn_wmma_f32_16x16x4_f32(false, a, false, b,
                                                               (short)0, acc[t],
                                                               false, false);
            }
        }
    }

    // ---- normalize and write: out layout (n*1024+p, h*80+d) ----
#pragma unroll
    for (int t = 0; t < 5; ++t)
#pragma unroll
        for (int i = 0; i < 8; ++i) {
            const int r = qRow + i + 8 * half;
            out[((size_t)n * MAX_PATCH + r) * DIM + h * HEAD_DIM + t * 16 + l16] =
                acc[t][i] / l_i[i];
        }
}

// ---------------------------------------------------------------------------
extern "C" void kernel_launch(void* const* d_in, const int* in_sizes, int n_in,
                              void* d_out, int out_size, void* d_ws, size_t ws_size,
                              hipStream_t stream) {
    const float* hidden = (const float*)d_in[0];  // (16,1024,1280)
    const float* rope   = (const float*)d_in[1];  // (16,1024,1,160)
    const int*   seq    = (const int*)d_in[2];    // (16,)
    const float* w_qkv  = (const float*)d_in[3];  // (3840,1280)
    const float* b_qkv  = (const float*)d_in[4];  // (3840,)
    const float* w_proj = (const float*)d_in[5];  // (1280,1280)
    const float* b_proj = (const float*)d_in[6];  // (1280,)
    float* out = (float*)d_out;                   // (16,1024,1280)

    float* qkv  = (float*)d_ws;                         // 16384 x 3840 f32
    float* attn = qkv + (size_t)ROWS * QKV_DIM;         // 16384 x 1280 f32

    // 1) QKV projection: 64x64 tiles with async-LDS double buffering.
    gemm_wmma_lds<<<dim3(QKV_DIM / BN, ROWS / BM), 256, 0, stream>>>(
        hidden, w_qkv, b_qkv, qkv, ROWS, QKV_DIM, DIM);

    // 2) RoPE on Q and K (in place).
    rope_kernel<<<ROWS, 256, 0, stream>>>(qkv, rope);

    // 3) Flash attention per (image, head, 64-row query block).
    flash_attn<<<dim3(MAX_PATCH / 64, NUM_HEADS, NUM_VISUAL), 128, 0, stream>>>(
        qkv, seq, attn);

    // 4) Output projection.
    gemm_wmma_lds<<<dim3(DIM / BN, ROWS / BM), 256, 0, stream>>>(
        attn, w_proj, b_proj, out, ROWS, DIM, DIM);
}